// Memory_58025008169465
// MI455X (gfx1250) — compile-verified
//
#include <hip/hip_runtime.h>
#include <math.h>

// ---------------------------------------------------------------------------
// MemAE-style memory module for MI455X (gfx1250, wave32).
// f32 GEMMs via V_WMMA_F32_16X16X4_F32, register-blocked 64x64 per wave
// (16 accumulators -> 16 WMMA per 8 fragment loads in the K loop).
// ---------------------------------------------------------------------------

typedef __attribute__((ext_vector_type(2))) float v2f;
typedef __attribute__((ext_vector_type(8))) float v8f;

constexpr int Bc  = 16;     // batch
constexpr int Cc  = 512;    // channels
constexpr int HWc = 1024;   // H*W = 32*32
constexpr int Mm  = 1024;   // memory slots
constexpr int Nn  = 16384;  // B*H*W queries

__device__ __forceinline__ v8f wmma_f32(v2f a, v2f b, v8f c) {
    // (neg_a, A, neg_b, B, c_mod, C, reuse_a, reuse_b)
    return __builtin_amdgcn_wmma_f32_16x16x4_f32(false, a, false, b,
                                                 (short)0, c, false, false);
}

// ---- f32 16x16x4 WMMA fragment layouts (wave32) ----------------------------
// A (16x4): lane L<16 -> row M=L, K={0,1}; L>=16 -> row M=L-16, K={2,3}.
// B (4x16): lane L<16 -> col N=L, K={0,1}; L>=16 -> col N=L-16, K={2,3}.
// C/D (16x16): lane L, VGPR v -> row M = v + 8*(L>=16), col N = L&15.

// ---------------------------------------------------------------------------
__global__ void k_init(float* qu, float* dl, float* sl, unsigned* emax) {
    int i = blockIdx.x * blockDim.x + threadIdx.x;
    if (i < Mm * Cc) qu[i] = 0.0f;
    if (i == 0) { *dl = 0.0f; *sl = 0.0f; *emax = 0u; }
}

// ‖keys[m]‖, one block per memory row
__global__ void k_keynorm(const float* __restrict__ keys, float* __restrict__ keynorm) {
    __shared__ float red[128];
    int m = blockIdx.x, tid = threadIdx.x;
    const float* row = keys + (size_t)m * Cc;
    float s = 0.0f;
    for (int c = tid; c < Cc; c += 128) { float v = row[c]; s += v * v; }
    red[tid] = s; __syncthreads();
    for (int off = 64; off > 0; off >>= 1) {
        if (tid < off) red[tid] += red[tid + off];
        __syncthreads();
    }
    if (tid == 0) keynorm[m] = sqrtf(red[0]);
}

// l2-normalize query over channels + BCHW->[N,C] transpose; also ‖qr[n]‖
__global__ void k_qnorm(const float* __restrict__ query,
                        float* __restrict__ qr, float* __restrict__ rownorm) {
    __shared__ float red[128];
    __shared__ float s_inv;
    int n = blockIdx.x, tid = threadIdx.x;
    int b = n >> 10, hw = n & (HWc - 1);
    const float* src = query + (size_t)b * Cc * HWc + hw;   // stride HWc over c
    float s = 0.0f;
    for (int c = tid; c < Cc; c += 128) { float v = src[(size_t)c * HWc]; s += v * v; }
    red[tid] = s; __syncthreads();
    for (int off = 64; off > 0; off >>= 1) {
        if (tid < off) red[tid] += red[tid + off];
        __syncthreads();
    }
    if (tid == 0) {
        float nrm = sqrtf(red[0]);
        float inv = 1.0f / fmaxf(nrm, 1e-12f);
        s_inv = inv;
        rownorm[n] = nrm * inv;   // ||qr[n]|| after normalization
    }
    __syncthreads();
    float inv = s_inv;
    float* dst = qr + (size_t)n * Cc;
    for (int c = tid; c < Cc; c += 128) dst[c] = src[(size_t)c * HWc] * inv;
}

// keysT[c, m] = keys[m, c] — LDS-tiled transpose, coalesced both sides
__global__ void k_transpose(const float* __restrict__ keys, float* __restrict__ keysT) {
    __shared__ float t[32][33];
    int tx = threadIdx.x, ty = threadIdx.y;      // (32, 8)
    int m0 = blockIdx.x * 32;
    int c0 = blockIdx.y * 32;
#pragma unroll
    for (int i = 0; i < 32; i += 8)
        t[ty + i][tx] = keys[(size_t)(m0 + ty + i) * Cc + c0 + tx];
    __syncthreads();
#pragma unroll
    for (int i = 0; i < 32; i += 8)
        keysT[(size_t)(c0 + ty + i) * Mm + m0 + tx] = t[tx][ty + i];
}

// max(entropy) via uint atomicMax (entropy > 0 so bit order == float order)
__global__ void k_entmax(const float* __restrict__ ent, unsigned* emax) {
    int i = blockIdx.x * blockDim.x + threadIdx.x;
    if (i < Nn) atomicMax(emax, __float_as_uint(ent[i]));
}

// C[i,j] = sum_k A[i,k]*B[j,k] (both K-contiguous).  64x64 tile per wave:
// 16 accumulators, 8 fragment loads feeding 16 WMMAs per K-step.
template <int K, int LDA, int LDB, int LDC>
__global__ void k_gemm_nt64(const float* __restrict__ A, const float* __restrict__ B,
                            float* __restrict__ Cmat) {
    int lane = threadIdx.x & 31;
    int wave = threadIdx.x >> 5;
    int row0 = blockIdx.y * 64;
    int col0 = (blockIdx.x * 4 + wave) * 64;
    int lr = lane & 15;
    int kh = (lane >> 4) << 1;
    const float* ap = A + (size_t)(row0 + lr) * LDA + kh;
    const float* bp = B + (size_t)(col0 + lr) * LDB + kh;
    v8f acc[4][4] = {};
#pragma unroll 2
    for (int k = 0; k < K; k += 4) {
        v2f a[4], b[4];
#pragma unroll
        for (int i = 0; i < 4; ++i) a[i] = *(const v2f*)(ap + (size_t)i * 16 * LDA + k);
#pragma unroll
        for (int j = 0; j < 4; ++j) b[j] = *(const v2f*)(bp + (size_t)j * 16 * LDB + k);
#pragma unroll
        for (int i = 0; i < 4; ++i)
#pragma unroll
            for (int j = 0; j < 4; ++j)
                acc[i][j] = wmma_f32(a[i], b[j], acc[i][j]);
    }
    int cn = lane & 15;
    int mo = (lane >> 4) * 8;
#pragma unroll
    for (int i = 0; i < 4; ++i)
#pragma unroll
        for (int j = 0; j < 4; ++j) {
            float* outp = Cmat + (size_t)(row0 + i * 16 + mo) * LDC + col0 + j * 16 + cn;
#pragma unroll
            for (int v = 0; v < 8; ++v) outp[(size_t)v * LDC] = acc[i][j][v];
        }
}

// per-column (over N) max and sum-of-exp; block owns 64 columns exclusively
__global__ void k_colstats(const float* __restrict__ score,
                           float* __restrict__ colmax, float* __restrict__ colsum) {
    __shared__ float red[4][64];
    __shared__ float cmx[64];
    int tx = threadIdx.x & 63, ty = threadIdx.x >> 6;
    int m = blockIdx.x * 64 + tx;
    float mx = -3.4e38f;
    for (int r = ty; r < Nn; r += 4) mx = fmaxf(mx, score[(size_t)r * Mm + m]);
    red[ty][tx] = mx; __syncthreads();
    if (ty == 0)
        cmx[tx] = fmaxf(fmaxf(red[0][tx], red[1][tx]), fmaxf(red[2][tx], red[3][tx]));
    __syncthreads();
    mx = cmx[tx];
    float s = 0.0f;
    for (int r = ty; r < Nn; r += 4) s += __expf(score[(size_t)r * Mm + m] - mx);
    red[ty][tx] = s; __syncthreads();
    if (ty == 0) {
        colmax[m] = mx;
        colsum[m] = red[0][tx] + red[1][tx] + red[2][tx] + red[3][tx];
    }
}

// fused per-row: argmax g, row softmax (sm, in place), column softmax (sq),
// similarity loss term, update weight w[n].
__global__ void k_rowsoftmax(float* __restrict__ score, float* __restrict__ sq,
                             const float* __restrict__ colmax,
                             const float* __restrict__ colsum,
                             const float* __restrict__ rownorm,
                             const float* __restrict__ keynorm,
                             const float* __restrict__ entropy,
                             const unsigned* __restrict__ entmaxbits,
                             int* __restrict__ argm, float* __restrict__ wvec,
                             float* __restrict__ sl) {
    __shared__ float smax[256];
    __shared__ int   simx[256];
    __shared__ float ssum[256];
    __shared__ float s_rmax, s_rsum;
    __shared__ int   s_g;
    int n = blockIdx.x, tid = threadIdx.x;
    float* row = score + (size_t)n * Mm;

    float mx = -3.4e38f; int mi = 0;
    for (int m = tid; m < Mm; m += 256) {
        float v = row[m];
        if (v > mx) { mx = v; mi = m; }
    }
    smax[tid] = mx; simx[tid] = mi; __syncthreads();
    for (int off = 128; off > 0; off >>= 1) {
        if (tid < off) {
            float v2 = smax[tid + off]; int i2 = simx[tid + off];
            if (v2 > smax[tid] || (v2 == smax[tid] && i2 < simx[tid])) {
                smax[tid] = v2; simx[tid] = i2;
            }
        }
        __syncthreads();
    }
    if (tid == 0) { s_rmax = smax[0]; s_g = simx[0]; }
    __syncthreads();
    float rmax = s_rmax;

    float s = 0.0f;
    for (int m = tid; m < Mm; m += 256) s += __expf(row[m] - rmax);
    ssum[tid] = s; __syncthreads();
    for (int off = 128; off > 0; off >>= 1) {
        if (tid < off) ssum[tid] += ssum[tid + off];
        __syncthreads();
    }
    if (tid == 0) s_rsum = ssum[0];
    __syncthreads();
    float rinv = 1.0f / s_rsum;

    for (int m = tid; m < Mm; m += 256) {
        float v = row[m];
        float smv = __expf(v - rmax) * rinv;
        float sqv = __expf(v - colmax[m]) / colsum[m];
        row[m] = smv;                       // in-place: raw score -> sm
        sq[(size_t)n * Mm + m] = sqv;
    }
    if (tid == 0) {
        int g = s_g;
        argm[n] = g;
        float ent = entropy[n];
        float emx = __uint_as_float(*entmaxbits);
        // cos = qr.keys[g] / max(||qr||*||keys[g]||, eps); numerator == rmax
        float cosv = rmax / fmaxf(rownorm[n] * keynorm[g], 1e-8f);
        atomicAdd(sl, (1.0f - cosv) * ent);
        // w = sq[n,g]/max_col(sq[:,g]) * ent/entmax == exp(rmax-colmax[g]) * ent/entmax
        wvec[n] = __expf(rmax - colmax[g]) * (ent / emx);
    }
}

// segment-sum scatter: qu[g[n], :] += w[n] * qr[n, :]
__global__ void k_scatter(const float* __restrict__ qr, const int* __restrict__ argm,
                          const float* __restrict__ wvec, float* __restrict__ qu) {
    int n = blockIdx.x;
    int g = argm[n];
    float w = wvec[n];
    float* dst = qu + (size_t)g * Cc;
    const float* src = qr + (size_t)n * Cc;
    for (int c = threadIdx.x; c < Cc; c += blockDim.x)
        atomicAdd(&dst[c], w * src[c]);
}

// updated_memory = l2norm(qu + keys)
__global__ void k_umem(const float* __restrict__ qu, const float* __restrict__ keys,
                       float* __restrict__ umem) {
    __shared__ float red[128];
    __shared__ float s_inv;
    int m = blockIdx.x, tid = threadIdx.x;
    const float* a = qu + (size_t)m * Cc;
    const float* b = keys + (size_t)m * Cc;
    float s = 0.0f;
    for (int c = tid; c < Cc; c += 128) { float v = a[c] + b[c]; s += v * v; }
    red[tid] = s; __syncthreads();
    for (int off = 64; off > 0; off >>= 1) {
        if (tid < off) red[tid] += red[tid + off];
        __syncthreads();
    }
    if (tid == 0) s_inv = 1.0f / fmaxf(sqrtf(red[0]), 1e-12f);
    __syncthreads();
    float inv = s_inv;
    float* dst = umem + (size_t)m * Cc;
    for (int c = tid; c < Cc; c += 128) dst[c] = (a[c] + b[c]) * inv;
}

// gram2 = U@U^T; 32x32 tile per wave (4 accs), fused off-diag^2 reduction
__global__ void k_gram_div(const float* __restrict__ U, float* __restrict__ dl) {
    int lane = threadIdx.x & 31;
    int wave = threadIdx.x >> 5;
    int row0 = blockIdx.y * 32;
    int col0 = (blockIdx.x * 4 + wave) * 32;
    int lr = lane & 15;
    int kh = (lane >> 4) << 1;
    const float* ap = U + (size_t)(row0 + lr) * Cc + kh;
    const float* bp = U + (size_t)(col0 + lr) * Cc + kh;
    v8f acc[2][2] = {};
#pragma unroll 4
    for (int k = 0; k < Cc; k += 4) {
        v2f a[2], b[2];
#pragma unroll
        for (int i = 0; i < 2; ++i) a[i] = *(const v2f*)(ap + (size_t)i * 16 * Cc + k);
#pragma unroll
        for (int j = 0; j < 2; ++j) b[j] = *(const v2f*)(bp + (size_t)j * 16 * Cc + k);
#pragma unroll
        for (int i = 0; i < 2; ++i)
#pragma unroll
            for (int j = 0; j < 2; ++j)
                acc[i][j] = wmma_f32(a[i], b[j], acc[i][j]);
    }
    int cn = lane & 15;
    int mo = (lane >> 4) * 8;
    float local = 0.0f;
#pragma unroll
    for (int i = 0; i < 2; ++i)
#pragma unroll
        for (int j = 0; j < 2; ++j) {
            int r0 = row0 + i * 16 + mo;
            int c1 = col0 + j * 16 + cn;
#pragma unroll
            for (int v = 0; v < 8; ++v) {
                float g = acc[i][j][v];
                if ((r0 + v) != c1) local += g * g;
            }
        }
    for (int off = 16; off > 0; off >>= 1)
        local += __shfl_down(local, off, 32);
    if (lane == 0)
        atomicAdd(dl, local * (1.0f / ((float)Mm * (float)Mm - (float)Mm)));
}

// updated_query[B,2C,H,W]: LDS-tiled transpose from [n,c] (qr / cm) to [c,hw]
__global__ void k_pack(const float* __restrict__ qr, const float* __restrict__ cm,
                       float* __restrict__ uq) {
    __shared__ float t[32][33];
    int tx = threadIdx.x, ty = threadIdx.y;      // (32, 8)
    int b  = blockIdx.z;
    int c0 = blockIdx.x * 32;                    // output channel tile [0, 2C)
    int hw0 = blockIdx.y * 32;
    const float* src = (c0 < Cc) ? qr : cm;
    int cc0 = (c0 < Cc) ? c0 : c0 - Cc;
#pragma unroll
    for (int i = 0; i < 32; i += 8) {
        int hw = hw0 + ty + i;
        t[ty + i][tx] = src[(size_t)(b * HWc + hw) * Cc + cc0 + tx];
    }
    __syncthreads();
    float* dst = uq + (size_t)b * 2 * Cc * HWc;
#pragma unroll
    for (int i = 0; i < 32; i += 8) {
        int ch = c0 + ty + i;
        dst[(size_t)ch * HWc + hw0 + tx] = t[tx][ty + i];
    }
}

// ---------------------------------------------------------------------------
extern "C" void kernel_launch(void* const* d_in, const int* in_sizes, int n_in,
                              void* d_out, int out_size, void* d_ws, size_t ws_size,
                              hipStream_t stream) {
    const float* query   = (const float*)d_in[0];   // [16,512,32,32]
    const float* keys    = (const float*)d_in[1];   // [1024,512]
    const float* entropy = (const float*)d_in[2];   // [16384,1]

    // --- output layout (flat, return order) ---
    float* uq   = (float*)d_out;                       // 16*1024*32*32
    float* umem = uq   + (size_t)Bc * 2 * Cc * HWc;    // 1024*512
    float* sqb  = umem + (size_t)Mm * Cc;              // 16384*1024
    float* smb  = sqb  + (size_t)Nn * Mm;              // 16384*1024 (raw score first)
    float* dl   = smb  + (size_t)Nn * Mm;
    float* sl   = dl + 1;

    // --- workspace carve-up (~72 MB) ---
    float* qr      = (float*)d_ws;             // N*C
    float* cm      = qr + (size_t)Nn * Cc;     // N*C
    float* qu      = cm + (size_t)Nn * Cc;     // M*C
    float* keysT   = qu + (size_t)Mm * Cc;     // C*M
    float* rownorm = keysT + (size_t)Cc * Mm;  // N
    float* wvec    = rownorm + Nn;             // N
    float* keynorm = wvec + Nn;                // M
    float* colmax  = keynorm + Mm;             // M
    float* colsum  = colmax + Mm;              // M
    int*   argm    = (int*)(colsum + Mm);      // N
    unsigned* emax = (unsigned*)(argm + Nn);   // 1

    k_init<<<(Mm * Cc + 255) / 256, 256, 0, stream>>>(qu, dl, sl, emax);
    k_keynorm<<<Mm, 128, 0, stream>>>(keys, keynorm);
    k_qnorm<<<Nn, 128, 0, stream>>>(query, qr, rownorm);
    k_transpose<<<dim3(Mm / 32, Cc / 32), dim3(32, 8), 0, stream>>>(keys, keysT);
    k_entmax<<<(Nn + 255) / 256, 256, 0, stream>>>(entropy, emax);

    // score = qr @ keys^T -> smb (raw logits staged in the sm output slot)
    k_gemm_nt64<Cc, Cc, Cc, Mm>
        <<<dim3(Mm / 256, Nn / 64), 128, 0, stream>>>(qr, keys, smb);
    k_colstats<<<Mm / 64, 256, 0, stream>>>(smb, colmax, colsum);
    k_rowsoftmax<<<Nn, 256, 0, stream>>>(smb, sqb, colmax, colsum, rownorm, keynorm,
                                         entropy, emax, argm, wvec, sl);

    // read: cm = sm @ keys  ==  sm @ keysT^T  (NT form, contiguous fragments)
    k_gemm_nt64<Mm, Mm, Mm, Cc>
        <<<dim3(Cc / 256, Nn / 64), 128, 0, stream>>>(smb, keysT, cm);

    // update: scatter, renormalize, diversity loss
    k_scatter<<<Nn, 256, 0, stream>>>(qr, argm, wvec, qu);
    k_umem<<<Mm, 128, 0, stream>>>(qu, keys, umem);
    k_gram_div<<<dim3(Mm / 128, Mm / 32), 128, 0, stream>>>(umem, dl);

    // updated_query pack (coalesced LDS-tile transpose)
    k_pack<<<dim3(2 * Cc / 32, HWc / 32, Bc), dim3(32, 8), 0, stream>>>(qr, cm, uq);
}